// GCN_25159918420052
// MI455X (gfx1250) — compile-verified
//
#include <hip/hip_runtime.h>

// ST-GCN fused spatial graph conv for MI455X (gfx1250, wave32).
//   f[n,k,c,t,v] = sum_ci x[n,ci,t,v] * W[k*64+c, ci] + b[k*64+c]
//   out[n,c,t,w] = sum_{k,v} f[n,k,c,t,v] * A[k,v,w]
// One workgroup (256 threads = 8 wave32) per (n, t).
// Both GEMMs done with V_WMMA_F32_16X16X4_F32.

#define N_    32
#define C_IN_ 64
#define T_    300
#define V_    25
#define K_    3
#define C_OUT_ 64
#define KC_   192          // K_*C_OUT_
#define KV_   75           // K_*V_  (aggregation K-dim)

typedef __attribute__((ext_vector_type(2))) float v2f;
typedef __attribute__((ext_vector_type(8))) float v8f;

#define XS_LD 36           // xs[ci][v-padded-32 + 4 pad]
#define FS_LD 196          // fs[v][o] with 4-col pad (192+4)

__global__ __launch_bounds__(256)
void stgcn_fused_kernel(const float* __restrict__ x,
                        const float* __restrict__ Wm,
                        const float* __restrict__ bias,
                        const float* __restrict__ Adj,
                        float* __restrict__ out)
{
    __shared__ float xs[C_IN_][XS_LD];   // x chunk, transposed: xs[ci][v], v padded to 32
    __shared__ float fs[32][FS_LD];      // f+bias chunk: fs[v][o], o = k*64+c
    __shared__ float As[KV_ * V_];       // adjacency as [75][25] row-major
    __shared__ float bs[KC_];            // bias

    const int tid  = threadIdx.x;
    const int lane = tid & 31;
    const int wave = tid >> 5;
    const int wg   = blockIdx.x;
    const int n    = wg / T_;
    const int t    = wg % T_;

    const int m    = lane & 15;          // M (or N) position within a 16-wide tile
    const int koff = (lane >> 4) << 1;   // K offset for A/B fragments: 0 or 2

    // ---------------- Phase 0: cooperative loads into LDS ----------------
    for (int i = tid; i < KV_ * V_; i += 256) As[i] = Adj[i];
    for (int i = tid; i < KC_;      i += 256) bs[i] = bias[i];

    // x[n, ci, t, v] -> xs[ci][v]; rows v=25..31 zero (GEMM padding).
    const float* xbase = x + (size_t)n * C_IN_ * (T_ * V_) + (size_t)t * V_;
    for (int i = tid; i < C_IN_ * 32; i += 256) {
        const int r  = i & 31;           // v (padded)
        const int ci = i >> 5;
        xs[ci][r] = (r < V_) ? xbase[(size_t)ci * (T_ * V_) + r] : 0.0f;
    }
    __syncthreads();

    // ---------------- Phase 1: projection GEMM (WMMA f32 16x16x4) --------
    // fs[v][o] = sum_ci xs[ci][v] * Wm[o*64 + ci] + bs[o]
    // M = v (2 tiles of 16, rows 25..31 padded), N = o (12 tiles), K = 64.
    for (int tile = wave; tile < 24; tile += 8) {
        const int mt  = tile & 1;
        const int nt  = tile >> 1;
        const int row = mt * 16 + m;                 // A-frag M index (v)
        const int ch  = nt * 16 + m;                 // B-frag N index (o)
        v8f acc = {0.f, 0.f, 0.f, 0.f, 0.f, 0.f, 0.f, 0.f};
        #pragma unroll
        for (int ks = 0; ks < 16; ++ks) {
            const int kb = ks * 4 + koff;            // K base for this lane half
            v2f a, bf;
            a.x  = xs[kb    ][row];
            a.y  = xs[kb + 1][row];
            const float* wp = Wm + (size_t)ch * C_IN_ + kb;  // W is [192,64] row-major
            bf.x = wp[0];
            bf.y = wp[1];
            acc = __builtin_amdgcn_wmma_f32_16x16x4_f32(
                      false, a, false, bf, (short)0, acc, false, false);
        }
        // C layout: VGPR r -> row = r + 8*(lane>=16), col = lane&15.
        const float bv = bs[ch];
        const int r0 = mt * 16 + ((lane >> 4) << 3);
        #pragma unroll
        for (int r = 0; r < 8; ++r)
            fs[r0 + r][ch] = acc[r] + bv;
    }
    __syncthreads();

    // ---------------- Phase 2: aggregation GEMM (WMMA f32 16x16x4) -------
    // out[c][w] = sum_{kv<75} fs[v][k*64+c] * As[kv*25 + w],  kv = k*25+v
    // M = c (4 tiles), N = w (2 tiles; cols >=25 masked), K = 75 pad-> 76.
    {
        const int mt = wave >> 1;                    // 0..3
        const int nt = wave & 1;                     // 0..1
        const int c  = mt * 16 + m;                  // A-frag M index (channel)
        const int w  = nt * 16 + m;                  // B-frag N index (graph node)
        const bool wok = (w < V_);
        v8f acc = {0.f, 0.f, 0.f, 0.f, 0.f, 0.f, 0.f, 0.f};
        #pragma unroll
        for (int ks = 0; ks < 19; ++ks) {
            const int kv0 = ks * 4 + koff;
            const int kv1 = kv0 + 1;
            v2f a, bf;
            // fs^T[c][kv] = fs[kv%25][(kv/25)*64 + c]; kv==75 is zero padding.
            if (kv0 < KV_) {
                const int k0 = kv0 / V_, v0 = kv0 - k0 * V_;
                a.x  = fs[v0][k0 * 64 + c];
                bf.x = wok ? As[kv0 * V_ + w] : 0.0f;
            } else { a.x = 0.0f; bf.x = 0.0f; }
            if (kv1 < KV_) {
                const int k1 = kv1 / V_, v1 = kv1 - k1 * V_;
                a.y  = fs[v1][k1 * 64 + c];
                bf.y = wok ? As[kv1 * V_ + w] : 0.0f;
            } else { a.y = 0.0f; bf.y = 0.0f; }
            acc = __builtin_amdgcn_wmma_f32_16x16x4_f32(
                      false, a, false, bf, (short)0, acc, false, false);
        }
        // Store C tile: row = c-tile row, col = w; skip padded w columns.
        if (wok) {
            const int rbase = mt * 16 + ((lane >> 4) << 3);
            float* op = out + ((size_t)n * C_OUT_) * (T_ * V_) + (size_t)t * V_ + w;
            #pragma unroll
            for (int r = 0; r < 8; ++r) {
                const int cc = rbase + r;
                op[(size_t)cc * (T_ * V_)] = acc[r];
            }
        }
    }
}

extern "C" void kernel_launch(void* const* d_in, const int* in_sizes, int n_in,
                              void* d_out, int out_size, void* d_ws, size_t ws_size,
                              hipStream_t stream) {
    const float* x   = (const float*)d_in[0];   // [32, 64, 300, 25]
    const float* Wm  = (const float*)d_in[1];   // [192, 64]
    const float* b   = (const float*)d_in[2];   // [192]
    const float* A   = (const float*)d_in[3];   // [3, 25, 25]
    float* out       = (float*)d_out;           // [32, 64, 300, 25]
    (void)in_sizes; (void)n_in; (void)out_size; (void)d_ws; (void)ws_size;

    dim3 grid(N_ * T_);   // 9600 workgroups, one per (n, t)
    dim3 block(256);      // 8 wave32
    hipLaunchKernelGGL(stgcn_fused_kernel, grid, block, 0, stream,
                       x, Wm, b, A, out);
}